// HashGrid_55645596287137
// MI455X (gfx1250) — compile-verified
//
#include <hip/hip_runtime.h>
#include <stdint.h>

// ---------------------------------------------------------------------------
// Hashed multi-LOD trilinear interpolation (Instant-NGP style) for MI455X.
//
// Strategy (see analysis): L2-resident gather kernel. LOD0 codebook (58,956 B)
// is DMA'd into LDS once per workgroup via the CDNA5 Tensor Data Mover
// (tensor_load_to_lds + s_wait_tensorcnt), then gathered from LDS; all other
// LODs gather from the 192MB L2. Fully unrolled 10x8 corner body for maximum
// memory-level parallelism. No WMMA: the workload has no matmul structure.
// ---------------------------------------------------------------------------

typedef unsigned int u32x4 __attribute__((ext_vector_type(4)));
typedef int          i32x4 __attribute__((ext_vector_type(4)));
typedef int          i32x8 __attribute__((ext_vector_type(8)));

#define LOD0_RES    17
#define LOD0_ENTRIES 4913                 // 17^3
#define LOD0_ELEMS  (3 * LOD0_ENTRIES)    // 14739 floats = 58,956 bytes

#if defined(__AMDGCN__) && __has_builtin(__builtin_amdgcn_tensor_load_to_lds)
#define USE_TDM 1
#else
#define USE_TDM 0
#endif

template <int RES, int GS, bool HASH, bool USE_LDS>
__device__ __forceinline__ void lod_accum(const float* __restrict__ gmem,
                                          const float* lmem,
                                          float px, float py, float pz,
                                          float& a0, float& a1, float& a2) {
  const float s  = 0.5f * (float)(RES - 1);
  const float cx = (px + 1.0f) * s;
  const float cy = (py + 1.0f) * s;
  const float cz = (pz + 1.0f) * s;
  const float hi = (float)(RES - 1) - 1e-5f;
  // floor of clamped coord (clamped value >= 0, so int-cast == floor)
  const int x0 = (int)floorf(fminf(fmaxf(cx, 0.0f), hi));
  const int y0 = (int)floorf(fminf(fmaxf(cy, 0.0f), hi));
  const int z0 = (int)floorf(fminf(fmaxf(cz, 0.0f), hi));
  const int x1 = min(x0 + 1, RES - 1);
  const int y1 = min(y0 + 1, RES - 1);
  const int z1 = min(z0 + 1, RES - 1);
  // weights use the UNCLAMPED continuous coordinate (matches reference)
  const float wx[2] = {(float)x1 - cx, cx - (float)x0};
  const float wy[2] = {(float)y1 - cy, cy - (float)y0};
  const float wz[2] = {(float)z1 - cz, cz - (float)z0};
  const int   xs[2] = {x0, x1};
  const int   ys[2] = {y0, y1};
  const int   zs[2] = {z0, z1};

  float r0 = 0.0f, r1 = 0.0f, r2 = 0.0f;
#pragma unroll
  for (int c = 0; c < 8; ++c) {
    // reference corner order: dx outer, dy, dz inner
    const int dx = (c >> 2) & 1;
    const int dy = (c >> 1) & 1;
    const int dz = c & 1;
    const int xi = xs[dx], yi = ys[dy], zi = zs[dz];
    unsigned idx;
    if (HASH) {
      // int32-wrapping multiply + xor; mod 2^19 == bitwise AND (two's compl.)
      const unsigned h = (unsigned)xi
                       ^ ((unsigned)yi * 265443567u)
                       ^ ((unsigned)zi * 805459861u);
      idx = h & (unsigned)(GS - 1);
    } else {
      idx = (unsigned)((zi * RES + yi) * RES + xi);
    }
    const float w = wx[dx] * wy[dy] * wz[dz];
    const float* p = (USE_LDS ? lmem : gmem) + 3u * idx;
    r0 = fmaf(w, p[0], r0);
    r1 = fmaf(w, p[1], r1);
    r2 = fmaf(w, p[2], r2);
  }
  // per-LOD ReLU, then accumulate (matches reference)
  a0 += fmaxf(r0, 0.0f);
  a1 += fmaxf(r1, 0.0f);
  a2 += fmaxf(r2, 0.0f);
}

__global__ void __launch_bounds__(256)
hashgrid_kernel(const float* __restrict__ pts,
                const float* __restrict__ cb0, const float* __restrict__ cb1,
                const float* __restrict__ cb2, const float* __restrict__ cb3,
                const float* __restrict__ cb4, const float* __restrict__ cb5,
                const float* __restrict__ cb6, const float* __restrict__ cb7,
                const float* __restrict__ cb8, const float* __restrict__ cb9,
                float* __restrict__ out, int n) {
  __shared__ float lds0[LOD0_ELEMS];  // 58,956 bytes: LOD0 codebook

#if USE_TDM
  // One TDM descriptor per workgroup: copy LOD0 codebook global -> LDS.
  // Issued by wave 0 only (TDM ignores EXEC; other waves never reach it).
  if (threadIdx.x < 32) {
    const uint64_t ga = (uint64_t)(uintptr_t)cb0;
    const uint32_t la = (uint32_t)(uintptr_t)lds0;  // low 32 bits = LDS offset
    u32x4 g0;
    g0[0] = 1u;                                           // count=1, user mode
    g0[1] = la;                                           // lds_addr
    g0[2] = (uint32_t)ga;                                 // global_addr[31:0]
    g0[3] = ((uint32_t)(ga >> 32) & 0x01FFFFFFu)          // global_addr[56:32]
            | 0x80000000u;                                // type=2 ("image")
    i32x8 g1;
    g1[0] = 0x00020000;                                   // data_size=2 (4B), no mask
    g1[1] = (int)((unsigned)LOD0_ELEMS << 16);            // tensor_dim0[15:0]
    g1[2] = (int)(((unsigned)LOD0_ELEMS >> 16) | (1u << 16)); // dim0 hi | tensor_dim1=1
    g1[3] = (int)((unsigned)LOD0_ELEMS << 16);            // tile_dim0 = 14739
    g1[4] = 1;                                            // tile_dim1=1, tile_dim2=0
    g1[5] = LOD0_ELEMS;                                   // tensor_dim0_stride lo
    g1[6] = 0;                                            // stride0 hi | stride1 lo
    g1[7] = 0;                                            // stride1 hi
    i32x4 g2 = {0, 0, 0, 0};
    i32x4 g3 = {0, 0, 0, 0};
#if defined(__clang_major__) && __clang_major__ >= 23
    i32x8 g4 = {0, 0, 0, 0, 0, 0, 0, 0};
    __builtin_amdgcn_tensor_load_to_lds(g0, g1, g2, g3, g4, 0);
#else
    __builtin_amdgcn_tensor_load_to_lds(g0, g1, g2, g3, 0);
#endif
#if __has_builtin(__builtin_amdgcn_s_wait_tensorcnt)
    __builtin_amdgcn_s_wait_tensorcnt(0);
#else
    asm volatile("s_wait_tensorcnt 0" ::: "memory");
#endif
  }
#else
  // Fallback: cooperative copy.
  for (int i = threadIdx.x; i < LOD0_ELEMS; i += (int)blockDim.x)
    lds0[i] = cb0[i];
#endif
  __syncthreads();

  const int stride = (int)(gridDim.x * blockDim.x);
  for (int i = (int)(blockIdx.x * blockDim.x + threadIdx.x); i < n; i += stride) {
    const float px = pts[3 * i + 0];
    const float py = pts[3 * i + 1];
    const float pz = pts[3 * i + 2];
    float a0 = 0.0f, a1 = 0.0f, a2 = 0.0f;
    // LODS = [17,24,35,51,75,110,162,238,349,513]; hash iff res^3 > 2^19
    lod_accum<17,    4913, false, true >(cb0, lds0, px, py, pz, a0, a1, a2);
    lod_accum<24,   13824, false, false>(cb1, lds0, px, py, pz, a0, a1, a2);
    lod_accum<35,   42875, false, false>(cb2, lds0, px, py, pz, a0, a1, a2);
    lod_accum<51,  132651, false, false>(cb3, lds0, px, py, pz, a0, a1, a2);
    lod_accum<75,  421875, false, false>(cb4, lds0, px, py, pz, a0, a1, a2);
    lod_accum<110, 524288, true,  false>(cb5, lds0, px, py, pz, a0, a1, a2);
    lod_accum<162, 524288, true,  false>(cb6, lds0, px, py, pz, a0, a1, a2);
    lod_accum<238, 524288, true,  false>(cb7, lds0, px, py, pz, a0, a1, a2);
    lod_accum<349, 524288, true,  false>(cb8, lds0, px, py, pz, a0, a1, a2);
    lod_accum<513, 524288, true,  false>(cb9, lds0, px, py, pz, a0, a1, a2);
    out[3 * i + 0] = a0;
    out[3 * i + 1] = a1;
    out[3 * i + 2] = a2;
  }
}

extern "C" void kernel_launch(void* const* d_in, const int* in_sizes, int n_in,
                              void* d_out, int out_size, void* d_ws, size_t ws_size,
                              hipStream_t stream) {
  (void)n_in; (void)d_ws; (void)ws_size; (void)out_size;
  const float* pts = (const float*)d_in[0];
  const float* cb0 = (const float*)d_in[1];
  const float* cb1 = (const float*)d_in[2];
  const float* cb2 = (const float*)d_in[3];
  const float* cb3 = (const float*)d_in[4];
  const float* cb4 = (const float*)d_in[5];
  const float* cb5 = (const float*)d_in[6];
  const float* cb6 = (const float*)d_in[7];
  const float* cb7 = (const float*)d_in[8];
  const float* cb8 = (const float*)d_in[9];
  const float* cb9 = (const float*)d_in[10];
  float* out = (float*)d_out;
  const int n = in_sizes[0] / 3;  // 2,097,152 points

  // 4096 blocks x 256 threads: 2 pts/thread via grid-stride loop; 59KB LDS per
  // block -> ~5 workgroups (40 wave32) per WGP, enough to hide L2 gather latency
  // while amortizing the TDM LDS fill.
  dim3 block(256);
  dim3 grid(4096);
  hashgrid_kernel<<<grid, block, 0, stream>>>(pts, cb0, cb1, cb2, cb3, cb4,
                                              cb5, cb6, cb7, cb8, cb9, out, n);
}